// SoftCrossEntropyLoss_87643102642550
// MI455X (gfx1250) — compile-verified
//
#include <hip/hip_runtime.h>

// Soft cross-entropy loss, B=64 S=2048 K=256 -> scalar mean.
// Memory-bound: ~268MB read once @ 23.3TB/s => ~11.5us floor.
// One wave32 per token, 2-token unroll for MLP (8 in-flight b128 NT loads),
// only 2 wave reductions per token (max, expsum); the target-weighted terms
// accumulate per-lane and are reduced once per wave at the end.

#define K_CLASSES 256
#define BLOCK_THREADS 256
#define WAVES_PER_BLOCK (BLOCK_THREADS / 32)
#define MAX_BLOCKS 1024

// Native clang vector (HIP float4 is a class type the NT builtin rejects).
typedef __attribute__((ext_vector_type(4))) float v4f;

__device__ __forceinline__ float wave_sum(float v) {
#pragma unroll
    for (int off = 16; off > 0; off >>= 1)
        v += __shfl_xor(v, off, 32);
    return v;
}

__device__ __forceinline__ float wave_max(float v) {
#pragma unroll
    for (int off = 16; off > 0; off >>= 1)
        v = fmaxf(v, __shfl_xor(v, off, 32));
    return v;
}

// Per-lane contribution of one token to the loss accumulator.
// x0/x1,t0/t1: this lane's 8 logits / 8 targets. mk: token mask.
__device__ __forceinline__ float token_contrib(v4f x0, v4f x1, v4f t0, v4f t1,
                                               float mk) {
    float m = fmaxf(fmaxf(fmaxf(x0.x, x0.y), fmaxf(x0.z, x0.w)),
                    fmaxf(fmaxf(x1.x, x1.y), fmaxf(x1.z, x1.w)));
    m = wave_max(m);

    float se = __expf(x0.x - m) + __expf(x0.y - m) +
               __expf(x0.z - m) + __expf(x0.w - m) +
               __expf(x1.x - m) + __expf(x1.y - m) +
               __expf(x1.z - m) + __expf(x1.w - m);
    se = wave_sum(se);
    float logZ = __logf(se) + m;                  // wave-uniform

    float st = (t0.x + t0.y) + (t0.z + t0.w) +
               (t1.x + t1.y) + (t1.z + t1.w);     // lane-local
    float stx = t0.x * x0.x + t0.y * x0.y + t0.z * x0.z + t0.w * x0.w +
                t1.x * x1.x + t1.y * x1.y + t1.z * x1.z + t1.w * x1.w;

    // loss_lane = mask * (logZ*sum_t - sum_tx); wave-summed only at the end.
    return mk * (logZ * st - stx);
}

__global__ void sce_partial_kernel(const float* __restrict__ inp,
                                   const float* __restrict__ targ,
                                   const float* __restrict__ mask,
                                   float* __restrict__ partial,
                                   int n_tok) {
    const int lane = threadIdx.x & 31;
    const int wave = threadIdx.x >> 5;
    const int gwave = blockIdx.x * WAVES_PER_BLOCK + wave;
    const int nwaves = gridDim.x * WAVES_PER_BLOCK;
    const int stride = 2 * nwaves;

    float acc = 0.0f;   // per-lane accumulator

    for (int tokA = gwave * 2; tokA < n_tok; tokA += stride) {
        const int hasB = (tokA + 1) < n_tok;
        const int tokB = hasB ? (tokA + 1) : tokA;   // clamped, safe address

        const v4f* ipa = (const v4f*)(inp + (size_t)tokA * K_CLASSES);
        const v4f* tpa = (const v4f*)(targ + (size_t)tokA * K_CLASSES);
        const v4f* ipb = (const v4f*)(inp + (size_t)tokB * K_CLASSES);
        const v4f* tpb = (const v4f*)(targ + (size_t)tokB * K_CLASSES);

        // Prefetch next pair (speculative; dropped if OOB). lane*8 floats =
        // 32B/lane => 32 lanes cover the full 1KB token A region per operand.
        const float* ip_next = inp + (size_t)(tokA + stride) * K_CLASSES + lane * 8;
        const float* tp_next = targ + (size_t)(tokA + stride) * K_CLASSES + lane * 8;
        __builtin_prefetch(ip_next, 0, 0);
        __builtin_prefetch(tp_next, 0, 0);

        // Issue all 8 NT b128 loads up front for max memory-level parallelism.
        v4f xa0 = __builtin_nontemporal_load(&ipa[lane]);
        v4f xa1 = __builtin_nontemporal_load(&ipa[lane + 32]);
        v4f ta0 = __builtin_nontemporal_load(&tpa[lane]);
        v4f ta1 = __builtin_nontemporal_load(&tpa[lane + 32]);
        v4f xb0 = __builtin_nontemporal_load(&ipb[lane]);
        v4f xb1 = __builtin_nontemporal_load(&ipb[lane + 32]);
        v4f tb0 = __builtin_nontemporal_load(&tpb[lane]);
        v4f tb1 = __builtin_nontemporal_load(&tpb[lane + 32]);

        const float mkA = mask[tokA];
        const float mkB = hasB ? mask[tokB] : 0.0f;

        acc += token_contrib(xa0, xa1, ta0, ta1, mkA);
        acc += token_contrib(xb0, xb1, tb0, tb1, mkB);
    }

    acc = wave_sum(acc);   // one reduction per wave for the linear terms

    __shared__ float smem[WAVES_PER_BLOCK];
    if (lane == 0) smem[wave] = acc;
    __syncthreads();
    if (threadIdx.x == 0) {
        float s = 0.0f;
#pragma unroll
        for (int w = 0; w < WAVES_PER_BLOCK; ++w) s += smem[w];
        partial[blockIdx.x] = s;
    }
}

__global__ void sce_finalize_kernel(const float* __restrict__ partial,
                                    int nblocks, float inv_n,
                                    float* __restrict__ out) {
    const int lane = threadIdx.x & 31;
    const int wave = threadIdx.x >> 5;

    float s = 0.0f;
    for (int i = threadIdx.x; i < nblocks; i += blockDim.x)
        s += partial[i];
    s = wave_sum(s);

    __shared__ float smem[WAVES_PER_BLOCK];
    if (lane == 0) smem[wave] = s;
    __syncthreads();
    if (threadIdx.x == 0) {
        float tot = 0.0f;
#pragma unroll
        for (int w = 0; w < WAVES_PER_BLOCK; ++w) tot += smem[w];
        out[0] = tot * inv_n;
    }
}

extern "C" void kernel_launch(void* const* d_in, const int* in_sizes, int n_in,
                              void* d_out, int out_size, void* d_ws, size_t ws_size,
                              hipStream_t stream) {
    const float* inp  = (const float*)d_in[0];   // [B,S,K] f32
    const float* targ = (const float*)d_in[1];   // [B,S,K] f32
    const float* mask = (const float*)d_in[2];   // [B,S]   f32
    float* out = (float*)d_out;

    const int n_tok = in_sizes[2];               // B*S = 131072

    int nblocks = MAX_BLOCKS;
    size_t need = (size_t)nblocks * sizeof(float);
    if (ws_size < need) {
        nblocks = (int)(ws_size / sizeof(float));
        if (nblocks < 1) nblocks = 1;
    }
    float* partial = (float*)d_ws;

    sce_partial_kernel<<<nblocks, BLOCK_THREADS, 0, stream>>>(
        inp, targ, mask, partial, n_tok);

    const float inv_n = 1.0f / (float)n_tok;
    sce_finalize_kernel<<<1, BLOCK_THREADS, 0, stream>>>(
        partial, nblocks, inv_n, out);
}